// GATLayer_74156905332795
// MI455X (gfx1250) — compile-verified
//
#include <hip/hip_runtime.h>
#include <math.h>

// ---------------------------------------------------------------------------
// GAT layer, MI455X (gfx1250), wave32. fp32 WMMA (V_WMMA_F32_16X16X4_F32) for
// both GEMMs; two-pass softmax so the [3,4096,4096] attention matrix is never
// materialized. Working set (~90MB) fits in the 192MB L2.
// ---------------------------------------------------------------------------

#define N_NODES 4096
#define DIN     512
#define DOUT    256
#define NHEAD   3
#define BN_EPS  1e-5f
#define NEG_BIG -1e30f

typedef float v2f __attribute__((ext_vector_type(2)));
typedef float v8f __attribute__((ext_vector_type(8)));

// workspace layout (float offsets)
#define WS_PROJ 0                                   // [H][N][DOUT]   12 MB
#define WS_SSRC (WS_PROJ + NHEAD * N_NODES * DOUT)  // [H][N]
#define WS_STGT (WS_SSRC + NHEAD * N_NODES)         // [H][N]
#define WS_RMAX (WS_STGT + NHEAD * N_NODES)         // [H][N]
#define WS_RSUM (WS_RMAX + NHEAD * N_NODES)         // [H][N]
#define WS_OPRE (WS_RSUM + NHEAD * N_NODES)         // [N][DOUT]      4 MB
#define WS_CSUM (WS_OPRE + N_NODES * DOUT)          // [DOUT]
#define WS_CSQ  (WS_CSUM + DOUT)                    // [DOUT]

// ---------------------------------------------------------------------------
// Kernel A: proj[h] = features @ W[h].  M=4096, K=512, N=256 per head.
// 8 waves/block as 2x4; each wave owns a 16x64 tile (4 f32 WMMA accumulators).
// ---------------------------------------------------------------------------
__global__ __launch_bounds__(256)
void gat_proj_kernel(const float* __restrict__ feat,
                     const float* __restrict__ weight,
                     float* __restrict__ ws) {
  const int h = blockIdx.z;
  const int blockRow = blockIdx.x * 32;
  const int tid = threadIdx.x;
  const int lane = tid & 31, wid = tid >> 5;
  const int rw = wid >> 2, cw = wid & 3;        // 2 row-waves x 4 col-waves
  const int r = lane & 15, kb = (lane >> 4) * 2;

  const float* Wh = weight + (size_t)h * DIN * DOUT;
  float* projH = ws + WS_PROJ + (size_t)h * N_NODES * DOUT;

  __shared__ float Alds[32 * 33];                 // 32 rows x 32 k, padded

  v8f acc[4];
#pragma unroll
  for (int t = 0; t < 4; ++t) {
#pragma unroll
    for (int e = 0; e < 8; ++e) acc[t][e] = 0.f;
  }

  for (int kc = 0; kc < DIN; kc += 32) {
    __syncthreads();
#pragma unroll
    for (int q = 0; q < 4; ++q) {                 // 1024 elems, coalesced
      int e = q * 256 + tid;
      int rr = e >> 5, kk = e & 31;
      Alds[rr * 33 + kk] = feat[(size_t)(blockRow + rr) * DIN + kc + kk];
    }
    __syncthreads();
#pragma unroll
    for (int kk = 0; kk < 32; kk += 4) {
      v2f a;                                      // A frag: 16x4 f32
      a.x = Alds[(rw * 16 + r) * 33 + kk + kb];
      a.y = Alds[(rw * 16 + r) * 33 + kk + kb + 1];
#pragma unroll
      for (int t = 0; t < 4; ++t) {
        int col = cw * 64 + t * 16 + r;
        v2f b;                                    // B frag: 4x16 f32
        b.x = Wh[(size_t)(kc + kk + kb) * DOUT + col];
        b.y = Wh[(size_t)(kc + kk + kb + 1) * DOUT + col];
        acc[t] = __builtin_amdgcn_wmma_f32_16x16x4_f32(
            false, a, false, b, (short)0, acc[t], false, false);
      }
    }
  }

  const int rowHi = (lane >> 4) << 3;             // lanes 16..31 hold M+8
#pragma unroll
  for (int t = 0; t < 4; ++t) {
    int col = cw * 64 + t * 16 + r;
#pragma unroll
    for (int e = 0; e < 8; ++e) {
      int row = blockRow + rw * 16 + e + rowHi;
      projH[(size_t)row * DOUT + col] = acc[t][e];
    }
  }
}

// ---------------------------------------------------------------------------
// Kernel B: s_src[h,n] = proj[h,n,:] . score_src[h];  same for s_tgt.
// One wave per (h,n); wave32 shuffle reduction.
// ---------------------------------------------------------------------------
__global__ __launch_bounds__(256)
void gat_scores_kernel(float* __restrict__ ws,
                       const float* __restrict__ score_src,
                       const float* __restrict__ score_tgt) {
  int task = blockIdx.x * 8 + (threadIdx.x >> 5);
  int lane = threadIdx.x & 31;
  int h = task / N_NODES, n = task % N_NODES;
  const float* p = ws + WS_PROJ + ((size_t)h * N_NODES + n) * DOUT;
  const float* ss = score_src + h * DOUT;
  const float* st = score_tgt + h * DOUT;
  float as = 0.f, at = 0.f;
  for (int e = lane; e < DOUT; e += 32) {
    float v = p[e];
    as += v * ss[e];
    at += v * st[e];
  }
#pragma unroll
  for (int off = 16; off > 0; off >>= 1) {
    as += __shfl_xor(as, off);
    at += __shfl_xor(at, off);
  }
  if (lane == 0) {
    ws[WS_SSRC + h * N_NODES + n] = as;
    ws[WS_STGT + h * N_NODES + n] = at;
  }
}

// ---------------------------------------------------------------------------
// Kernel C: softmax pass 1. One wave per row i; one coalesced adj sweep feeds
// online (max,sum) for all 3 heads. Sentinel -1e30 avoids (-inf - -inf) NaN.
// ---------------------------------------------------------------------------
__global__ __launch_bounds__(256)
void gat_stats_kernel(const float* __restrict__ adj, float* __restrict__ ws) {
  int i = blockIdx.x * 8 + (threadIdx.x >> 5);
  int lane = threadIdx.x & 31;
  float ssrc[NHEAD], m[NHEAD], s[NHEAD];
#pragma unroll
  for (int h = 0; h < NHEAD; ++h) {
    ssrc[h] = ws[WS_SSRC + h * N_NODES + i];
    m[h] = NEG_BIG;
    s[h] = 0.f;
  }
  const float* arow = adj + (size_t)i * N_NODES;
  for (int j = lane; j < N_NODES; j += 32) {
    float a = arow[j];
    if (j + 512 < N_NODES) __builtin_prefetch(&arow[j + 512], 0, 1);
    if (a != 0.f) {
#pragma unroll
      for (int h = 0; h < NHEAD; ++h) {
        float x = ssrc[h] + ws[WS_STGT + h * N_NODES + j];
        x = (x > 0.f) ? x : 0.2f * x;              // leaky_relu(0.2)
        x += a;                                    // conn adds edge weight
        float nm = fmaxf(m[h], x);
        s[h] = s[h] * __expf(m[h] - nm) + __expf(x - nm);
        m[h] = nm;
      }
    }
  }
#pragma unroll
  for (int off = 16; off > 0; off >>= 1) {
#pragma unroll
    for (int h = 0; h < NHEAD; ++h) {
      float om = __shfl_xor(m[h], off);
      float os = __shfl_xor(s[h], off);
      float nm = fmaxf(m[h], om);
      s[h] = s[h] * __expf(m[h] - nm) + os * __expf(om - nm);
      m[h] = nm;
    }
  }
  if (lane == 0) {
#pragma unroll
    for (int h = 0; h < NHEAD; ++h) {
      ws[WS_RMAX + h * N_NODES + i] = m[h];
      ws[WS_RSUM + h * N_NODES + i] = s[h];
    }
  }
}

// ---------------------------------------------------------------------------
// Kernel D: softmax pass 2 fused with att @ proj (+head-mean +bias).
// Block = 32-row i-tile, 8 waves (2x4), wave tile 16x64, acc stays in VGPRs
// across the whole j loop and all heads. P tile (32x32, already normalized
// by 1/rowsum) built cooperatively in LDS, then consumed by f32 WMMA.
// proj re-reads hit the 192MB L2.
// ---------------------------------------------------------------------------
__global__ __launch_bounds__(256)
void gat_attn_kernel(const float* __restrict__ adj,
                     const float* __restrict__ bias,
                     float* __restrict__ ws) {
  const int iTile = blockIdx.x * 32;
  const int tid = threadIdx.x;
  const int lane = tid & 31, wid = tid >> 5;
  const int rw = wid >> 2, cw = wid & 3;
  const int r = lane & 15, kb = (lane >> 4) * 2;

  __shared__ float P[32 * 33];
  __shared__ float rowM[32], rowIS[32], rowSrc[32];

  v8f acc[4];
#pragma unroll
  for (int t = 0; t < 4; ++t) {
#pragma unroll
    for (int e = 0; e < 8; ++e) acc[t][e] = 0.f;
  }

  for (int h = 0; h < NHEAD; ++h) {
    const float* projH = ws + WS_PROJ + (size_t)h * N_NODES * DOUT;
    const float* stgt = ws + WS_STGT + h * N_NODES;
    __syncthreads();                       // prior head's P readers done
    if (tid < 32) {
      int i = iTile + tid;
      rowM[tid] = ws[WS_RMAX + h * N_NODES + i];
      rowIS[tid] = 1.f / ws[WS_RSUM + h * N_NODES + i];
      rowSrc[tid] = ws[WS_SSRC + h * N_NODES + i];
    }
    __syncthreads();

    for (int jc = 0; jc < N_NODES; jc += 32) {
#pragma unroll
      for (int q = 0; q < 4; ++q) {        // 32x32 tile, coalesced adj rows
        int e = q * 256 + tid;
        int il = e >> 5, jl = e & 31;
        int j = jc + jl;
        float a = adj[(size_t)(iTile + il) * N_NODES + j];
        float p = 0.f;
        if (a != 0.f) {
          float x = rowSrc[il] + stgt[j];
          x = (x > 0.f) ? x : 0.2f * x;
          p = __expf(x + a - rowM[il]) * rowIS[il];   // normalized attention
        }
        P[il * 33 + jl] = p;
      }
      if (tid < 32 && jc + 32 < N_NODES)   // stream next adj tile into cache
        __builtin_prefetch(&adj[(size_t)(iTile + tid) * N_NODES + jc + 32], 0, 1);
      __syncthreads();                     // P ready (EXEC all-1 again here)
#pragma unroll
      for (int kk = 0; kk < 32; kk += 4) {
        v2f a;                             // A frag from LDS P tile
        a.x = P[(rw * 16 + r) * 33 + kk + kb];
        a.y = P[(rw * 16 + r) * 33 + kk + kb + 1];
#pragma unroll
        for (int t = 0; t < 4; ++t) {
          int col = cw * 64 + t * 16 + r;
          v2f b;                           // B frag from proj (L2-resident)
          b.x = projH[(size_t)(jc + kk + kb) * DOUT + col];
          b.y = projH[(size_t)(jc + kk + kb + 1) * DOUT + col];
          acc[t] = __builtin_amdgcn_wmma_f32_16x16x4_f32(
              false, a, false, b, (short)0, acc[t], false, false);
        }
      }
      __syncthreads();                     // before P is overwritten
    }
  }

  const int rowHi = (lane >> 4) << 3;
  const float invH = 1.f / (float)NHEAD;
  float* opre = ws + WS_OPRE;
#pragma unroll
  for (int t = 0; t < 4; ++t) {
    int col = cw * 64 + t * 16 + r;
    float bv = bias[col];
#pragma unroll
    for (int e = 0; e < 8; ++e) {
      int row = iTile + rw * 16 + e + rowHi;
      opre[(size_t)row * DOUT + col] = acc[t][e] * invH + bv;
    }
  }
}

// ---------------------------------------------------------------------------
// Kernel E1: per-column sum / sumsq over the node dim (atomics, coalesced).
// ---------------------------------------------------------------------------
__global__ __launch_bounds__(256)
void gat_bnstats_kernel(float* __restrict__ ws) {
  int c = threadIdx.x;
  int r0 = blockIdx.x * 64;
  const float* opre = ws + WS_OPRE;
  float s = 0.f, sq = 0.f;
  for (int k = 0; k < 64; ++k) {
    float v = opre[(size_t)(r0 + k) * DOUT + c];
    s += v;
    sq += v * v;
  }
  atomicAdd(&ws[WS_CSUM + c], s);
  atomicAdd(&ws[WS_CSQ + c], sq);
}

// ---------------------------------------------------------------------------
// Kernel E2: batchnorm (biased var) + relu -> d_out.
// ---------------------------------------------------------------------------
__global__ __launch_bounds__(256)
void gat_bnapply_kernel(const float* __restrict__ gamma,
                        const float* __restrict__ beta,
                        const float* __restrict__ ws,
                        float* __restrict__ out) {
  int idx0 = blockIdx.x * 256 + threadIdx.x;
  for (int idx = idx0; idx < N_NODES * DOUT; idx += 1024 * 256) {
    int c = idx & (DOUT - 1);
    float mu = ws[WS_CSUM + c] * (1.f / N_NODES);
    float var = ws[WS_CSQ + c] * (1.f / N_NODES) - mu * mu;
    float x = ws[WS_OPRE + idx];
    float y = (x - mu) * rsqrtf(var + BN_EPS) * gamma[c] + beta[c];
    out[idx] = fmaxf(y, 0.f);
  }
}

// ---------------------------------------------------------------------------
extern "C" void kernel_launch(void* const* d_in, const int* in_sizes, int n_in,
                              void* d_out, int out_size, void* d_ws, size_t ws_size,
                              hipStream_t stream) {
  (void)in_sizes; (void)n_in; (void)out_size; (void)ws_size;
  const float* feat   = (const float*)d_in[0];
  const float* adj    = (const float*)d_in[1];
  const float* weight = (const float*)d_in[2];
  const float* bias   = (const float*)d_in[3];
  const float* ssrc   = (const float*)d_in[4];
  const float* stgt   = (const float*)d_in[5];
  const float* gamma  = (const float*)d_in[6];
  const float* beta   = (const float*)d_in[7];
  float* ws  = (float*)d_ws;
  float* out = (float*)d_out;

  gat_proj_kernel<<<dim3(N_NODES / 32, 1, NHEAD), 256, 0, stream>>>(feat, weight, ws);
  gat_scores_kernel<<<(NHEAD * N_NODES) / 8, 256, 0, stream>>>(ws, ssrc, stgt);
  gat_stats_kernel<<<N_NODES / 8, 256, 0, stream>>>(adj, ws);
  gat_attn_kernel<<<N_NODES / 32, 256, 0, stream>>>(adj, bias, ws);
  hipMemsetAsync((void*)(ws + WS_CSUM), 0, 2 * DOUT * sizeof(float), stream);
  gat_bnstats_kernel<<<N_NODES / 64, 256, 0, stream>>>(ws);
  gat_bnapply_kernel<<<1024, 256, 0, stream>>>(gamma, beta, ws, out);
}